// CliffordLayerNorm_13950053777614
// MI455X (gfx1250) — compile-verified
//
#include <hip/hip_runtime.h>

// ---------------------------------------------------------------------------
// CliffordLayerNorm for MI455X (gfx1250, wave32).
//
// Roofline: 256 MiB read + 256 MiB write @ 23.3 TB/s => ~23 us floor,
// ~0.8 GFLOP total (1.5 flop/byte) => pure bandwidth problem; WMMA cannot
// help (one-hot einsum == 9-bin segmented reduction, and matrix-unit layouts
// would add cross-lane permutes without reducing bytes). Strategy:
//   * one wave32 per 256-float block,
//   * async global->LDS DMA (ASYNCcnt), 3-buffer / distance-2 prefetch,
//   * grade reduction via popcount separability: grade(8l+j)=popc(l)+popc(j),
//     so each lane only accumulates 4 grade slots; cross-lane merge with
//     per-wave LDS ds_add_f32 atomics (in-order within a wave, no barrier),
//   * single pass: var = E[x^2] - mean^2,
//   * loop-invariant per-lane 1/count, weight, bias window,
//   * 128-bit nontemporal stores (512 MB stream > 192 MB L2, no reuse).
// ---------------------------------------------------------------------------

typedef float v4f __attribute__((ext_vector_type(4)));

#define WAVES_PER_WG 8
#define NBUF 3

// C(8, g) for g = 0..8
__device__ const float kCnt[9] = {1.f, 8.f, 28.f, 56.f, 70.f, 56.f, 28.f, 8.f, 1.f};

__global__ __launch_bounds__(256) void clifford_ln_kernel(
    const float* __restrict__ x, const float* __restrict__ w,
    const float* __restrict__ bs, float* __restrict__ out, int nblocks)
{
    // Per-wave triple buffer for async DMA + per-wave 32-slot stats table
    // (slots [0..8] = grade sums, [16..24] = grade sums of squares).
    __shared__ __align__(16) float xbuf[WAVES_PER_WG][NBUF][256];
    __shared__ float stats[WAVES_PER_WG][32];

    const int lane = threadIdx.x & 31;
    const int wid  = threadIdx.x >> 5;
    const int gw   = blockIdx.x * WAVES_PER_WG + wid;   // global wave id
    const int nw   = gridDim.x * WAVES_PER_WG;          // total waves
    const int base = __popc(lane);                      // 0..5

    // Loop-invariant per-lane affine window for grades base..base+3.
    float ic[4], wj[4], bj[4];
#pragma unroll
    for (int j = 0; j < 4; ++j) {
        ic[j] = 1.0f / kCnt[base + j];
        wj[j] = w[base + j];
        bj[j] = bs[base + j];
    }

    // Numeric LDS byte address of this lane's 16B slice in buffer 0.
    // (low 32 bits of a generic pointer to LDS == LDS byte offset)
    const unsigned ldsbase = (unsigned)(unsigned long long)&xbuf[wid][0][lane * 4];

    // Async DMA of one 1 KiB block: 2 x b128 per lane; the instruction
    // offset is added to BOTH the global and the LDS address.
    auto prefetch = [&](int blk, unsigned lds) {
        const unsigned goff = (unsigned)blk * 1024u + (unsigned)lane * 16u;
        asm volatile(
            "global_load_async_to_lds_b128 %0, %1, %2\n\t"
            "global_load_async_to_lds_b128 %0, %1, %2 offset:512"
            :: "v"(lds), "v"(goff), "s"(x) : "memory");
    };

    int b  = gw;
    int pb = 0;

    // Prime a distance-2 pipeline.
    if (b < nblocks)      prefetch(b,      ldsbase);
    if (b + nw < nblocks) prefetch(b + nw, ldsbase + 1024u);

    for (; b < nblocks; b += nw) {
        // Zero this wave's stats table early (overlaps the DMA wait;
        // LDS ops from one wave are in-order, so it is safely after the
        // previous iteration's stats reads and before this one's atomics).
        stats[wid][lane] = 0.0f;

        const int b2 = b + 2 * nw;
        if (b2 < nblocks) {
            const int nb = (pb + 2 >= NBUF) ? pb + 2 - NBUF : pb + 2;
            prefetch(b2, ldsbase + (unsigned)nb * 1024u);
            asm volatile("s_wait_asynccnt 4" ::: "memory");  // current landed
        } else if (b + nw < nblocks) {
            asm volatile("s_wait_asynccnt 2" ::: "memory");
        } else {
            asm volatile("s_wait_asynccnt 0" ::: "memory");
        }

        // Lane owns elements m = 8*lane + j, j = 0..7 (two ds_load_b128).
        const v4f* lx = (const v4f*)&xbuf[wid][pb][lane * 8];
        const v4f a = lx[0];   // j = 0..3
        const v4f c = lx[1];   // j = 4..7

        // popc(j) for j=0..7 is {0,1,1,2,1,2,2,3}: 4 grade slots per lane.
        const float s0 = a.x;
        const float s1 = a.y + a.z + c.x;
        const float s2 = a.w + c.y + c.z;
        const float s3 = c.w;
        const float q0 = a.x * a.x;
        const float q1 = fmaf(a.y, a.y, fmaf(a.z, a.z, c.x * c.x));
        const float q2 = fmaf(a.w, a.w, fmaf(c.y, c.y, c.z * c.z));
        const float q3 = c.w * c.w;

        // Wave-private 9-bin reduction (ds_add_f32, no return). Within one
        // atomic instruction lanes with equal popcount collide (<=10-way).
        atomicAdd(&stats[wid][base + 0], s0);
        atomicAdd(&stats[wid][base + 1], s1);
        atomicAdd(&stats[wid][base + 2], s2);
        atomicAdd(&stats[wid][base + 3], s3);
        atomicAdd(&stats[wid][16 + base + 0], q0);
        atomicAdd(&stats[wid][16 + base + 1], q1);
        atomicAdd(&stats[wid][16 + base + 2], q2);
        atomicAdd(&stats[wid][16 + base + 3], q3);
        __builtin_amdgcn_wave_barrier();   // scheduling fence only

        float scale[4], off[4];
#pragma unroll
        for (int j = 0; j < 4; ++j) {
            const float m   = stats[wid][base + j] * ic[j];
            const float var = fmaf(stats[wid][16 + base + j], ic[j], -m * m);
            const float inv = rsqrtf(var + 1e-5f);
            scale[j] = inv * wj[j];
            off[j]   = fmaf(-m, scale[j], bj[j]);   // bias - mean*scale
        }

        // out = x*scale_g + (bias_g - mean_g*scale_g); slot index = popc(j).
        v4f o0, o1;
        o0.x = fmaf(a.x, scale[0], off[0]);
        o0.y = fmaf(a.y, scale[1], off[1]);
        o0.z = fmaf(a.z, scale[1], off[1]);
        o0.w = fmaf(a.w, scale[2], off[2]);
        o1.x = fmaf(c.x, scale[1], off[1]);
        o1.y = fmaf(c.y, scale[2], off[2]);
        o1.z = fmaf(c.z, scale[2], off[2]);
        o1.w = fmaf(c.w, scale[3], off[3]);

        float* op = out + (size_t)b * 256 + lane * 8;
        __builtin_nontemporal_store(o0, (v4f*)op);
        __builtin_nontemporal_store(o1, (v4f*)(op + 4));

        pb = (pb + 1 >= NBUF) ? 0 : pb + 1;
    }
}

extern "C" void kernel_launch(void* const* d_in, const int* in_sizes, int n_in,
                              void* d_out, int out_size, void* d_ws, size_t ws_size,
                              hipStream_t stream) {
    const float* x  = (const float*)d_in[0];   // [16, 4096, 1024] f32
    const float* w  = (const float*)d_in[1];   // [9] f32
    const float* bs = (const float*)d_in[2];   // [9] f32
    float* out = (float*)d_out;

    const int nblocks = in_sizes[0] / 256;     // 262144 independent blocks
    int grid = (nblocks + WAVES_PER_WG - 1) / WAVES_PER_WG;
    if (grid > 4096) grid = 4096;              // persistent waves, grid-stride
    if (grid < 1) grid = 1;

    clifford_ln_kernel<<<grid, 256, 0, stream>>>(x, w, bs, out, nblocks);
}